// FocalBoxLoss_87462714016414
// MI455X (gfx1250) — compile-verified
//
#include <hip/hip_runtime.h>
#include <hip/hip_bf16.h>

// Problem constants (from reference)
#define B_N 16
#define P_N 2304
#define G_N 64
#define C_N 512
#define NBLK 4096   // partial-sum blocks for focal loss (64*64, matches WMMA tiling)

typedef __attribute__((ext_vector_type(2))) float v2f;
typedef __attribute__((ext_vector_type(8))) float v8f;

// ---------------------------------------------------------------------------
// Kernel A: cost[b][g][p] = 1 - IoU(pred_boxes[b][p], boxes[b][g])  (float64,
// matching the host-side numpy float64 round-trip in the reference)
// ---------------------------------------------------------------------------
__global__ void cost_kernel(const float* __restrict__ pb,
                            const float* __restrict__ gb,
                            double* __restrict__ cost)
{
    int e = blockIdx.x * blockDim.x + threadIdx.x;
    const int total = B_N * G_N * P_N;
    if (e >= total) return;
    int p = e % P_N;
    int g = (e / P_N) % G_N;
    int b = e / (P_N * G_N);
    const float* A  = pb + ((size_t)b * P_N + p) * 4;
    const float* Bx = gb + ((size_t)b * G_N + g) * 4;
    double a0 = A[0],  a1 = A[1],  a2 = A[2],  a3 = A[3];
    double b0 = Bx[0], b1 = Bx[1], b2 = Bx[2], b3 = Bx[3];
    double aa = (a2 - a0) * (a3 - a1);
    double ab = (b2 - b0) * (b3 - b1);
    double ltx = fmax(a0, b0), lty = fmax(a1, b1);
    double rbx = fmin(a2, b2), rby = fmin(a3, b3);
    double w = fmax(rbx - ltx, 0.0), h = fmax(rby - lty, 0.0);
    double inter = w * h;
    double iou = inter / (aa + ab - inter);
    cost[e] = 1.0 - iou;     // e == (b*G + g)*P + p : row = gt, col = pred
}

// ---------------------------------------------------------------------------
// Kernel B: Jonker-Volgenant shortest augmenting path, one block per image.
// n = G_N rows (gt), m = P_N cols (pred).  All state in LDS, double precision.
// ---------------------------------------------------------------------------
__global__ void hungarian_kernel(const double* __restrict__ cost,
                                 int* __restrict__ lsa,
                                 int* __restrict__ idxArr)
{
    const int M = P_N, N = G_N;
    __shared__ double s_v[P_N + 1];
    __shared__ double s_minv[P_N + 1];
    __shared__ double s_u[G_N + 1];
    __shared__ double s_rv[256];
    __shared__ int    s_p[P_N + 1];
    __shared__ int    s_way[P_N + 1];
    __shared__ int    s_ri[256];
    __shared__ int    s_usedList[G_N + 2];
    __shared__ int    s_g2p[G_N];
    __shared__ unsigned char s_used[P_N + 1];
    __shared__ int    s_j0, s_i0v, s_usedCnt;

    const int b   = blockIdx.x;
    const int tid = threadIdx.x;
    const int bd  = blockDim.x;
    const double* cb = cost + (size_t)b * G_N * P_N;
    const double INF = __builtin_inf();

    for (int j = tid; j <= M; j += bd) { s_v[j] = 0.0; s_p[j] = 0; s_way[j] = 0; }
    for (int i = tid; i <= N; i += bd) s_u[i] = 0.0;
    __syncthreads();

    for (int i = 1; i <= N; ++i) {
        for (int j = tid; j <= M; j += bd) { s_minv[j] = INF; s_used[j] = 0; }
        if (tid == 0) { s_p[0] = i; s_j0 = 0; s_usedCnt = 0; }
        __syncthreads();

        while (true) {
            if (tid == 0) {
                int j0 = s_j0;
                s_used[j0] = 1;
                s_usedList[s_usedCnt++] = j0;
                s_i0v = s_p[j0];
            }
            __syncthreads();
            const int    i0  = s_i0v;
            const int    j0c = s_j0;
            const double ui0 = s_u[i0];
            const double* crow = cb + (size_t)(i0 - 1) * P_N;
            for (int j = 1 + tid; j <= M; j += bd) {
                if (!s_used[j]) {
                    double cur = crow[j - 1] - ui0 - s_v[j];
                    if (cur < s_minv[j]) { s_minv[j] = cur; s_way[j] = j0c; }
                }
            }
            __syncthreads();
            // argmin over free columns (first-min tie-break like np.argmin)
            double best = INF; int bidx = M + 1;
            for (int j = 1 + tid; j <= M; j += bd) {
                if (!s_used[j]) {
                    double mv = s_minv[j];
                    if (mv < best || (mv == best && j < bidx)) { best = mv; bidx = j; }
                }
            }
            s_rv[tid] = best; s_ri[tid] = bidx;
            __syncthreads();
            for (int s = bd >> 1; s > 0; s >>= 1) {
                if (tid < s) {
                    double ov = s_rv[tid + s]; int oi = s_ri[tid + s];
                    if (ov < s_rv[tid] || (ov == s_rv[tid] && oi < s_ri[tid])) {
                        s_rv[tid] = ov; s_ri[tid] = oi;
                    }
                }
                __syncthreads();
            }
            const double delta = s_rv[0];
            const int    j1    = s_ri[0];
            // u[p[used]] += delta ; v[used] -= delta  (rows p[jj] are distinct)
            for (int k = tid; k < s_usedCnt; k += bd) {
                int jj = s_usedList[k];
                s_u[s_p[jj]] += delta;
                s_v[jj]      -= delta;
            }
            for (int j = 1 + tid; j <= M; j += bd)
                if (!s_used[j]) s_minv[j] -= delta;
            __syncthreads();
            if (tid == 0) s_j0 = j1;
            if (s_p[j1] == 0) break;    // uniform: all threads share j1
        }
        // augment (serial, thread 0)
        if (tid == 0) {
            int j0 = s_j0;
            while (j0 != 0) { int j1 = s_way[j0]; s_p[j0] = s_p[j1]; j0 = j1; }
        }
        __syncthreads();
    }

    // row_to_col (g2p), then rank-sort (distinct values -> argsort equivalent)
    for (int j = 1 + tid; j <= M; j += bd) {
        int pi = s_p[j];
        if (pi != 0) s_g2p[pi - 1] = j - 1;
    }
    __syncthreads();
    if (tid < N) {
        int val  = s_g2p[tid];
        int rank = 0;
        for (int g = 0; g < N; ++g) rank += (s_g2p[g] < val);
        lsa[b * G_N + rank]    = val;   // sorted pred indices
        idxArr[b * G_N + rank] = tid;   // gt permutation
    }
}

// ---------------------------------------------------------------------------
// Kernel C1: write pm = pred_boxes[b, lsa] for all b; box loss for b = B-1
// ---------------------------------------------------------------------------
__global__ void boxloss_pm_kernel(const float* __restrict__ pred_boxes,
                                  const float* __restrict__ boxes,
                                  const int* __restrict__ lsa,
                                  const int* __restrict__ idxArr,
                                  float* __restrict__ out)
{
    __shared__ float red[G_N];
    int b = blockIdx.x, g = threadIdx.x;
    int l  = lsa[b * G_N + g];
    int ix = idxArr[b * G_N + g];
    const float* pmp = pred_boxes + ((size_t)b * P_N + l) * 4;
    float pm0 = pmp[0], pm1 = pmp[1], pm2 = pmp[2], pm3 = pmp[3];
    float* od = out + 2 + ((size_t)b * G_N + g) * 4;
    od[0] = pm0; od[1] = pm1; od[2] = pm2; od[3] = pm3;

    if (b == B_N - 1) {
        const float* gmp = boxes + ((size_t)b * G_N + ix) * 4;
        float gm0 = gmp[0], gm1 = gmp[1], gm2 = gmp[2], gm3 = gmp[3];
        float loss = 0.f;
        float d;
        d = fabsf(pm0 - gm0); loss += (d < 1.f) ? 0.5f * d * d : d - 0.5f;
        d = fabsf(pm1 - gm1); loss += (d < 1.f) ? 0.5f * d * d : d - 0.5f;
        d = fabsf(pm2 - gm2); loss += (d < 1.f) ? 0.5f * d * d : d - 0.5f;
        d = fabsf(pm3 - gm3); loss += (d < 1.f) ? 0.5f * d * d : d - 0.5f;
        float aa = (pm2 - pm0) * (pm3 - pm1);
        float ab = (gm2 - gm0) * (gm3 - gm1);
        float ltx = fmaxf(pm0, gm0), lty = fmaxf(pm1, gm1);
        float rbx = fminf(pm2, gm2), rby = fminf(pm3, gm3);
        float w = fmaxf(rbx - ltx, 0.f), h = fmaxf(rby - lty, 0.f);
        float inter = w * h;
        float iou = inter / (aa + ab - inter);
        loss += 1.f - iou;
        red[g] = loss;
        __syncthreads();
        for (int s = G_N >> 1; s > 0; s >>= 1) {
            if (g < s) red[g] += red[g + s];
            __syncthreads();
        }
        if (g == 0) out[0] = red[0] / (float)B_N;
    }
}

// ---------------------------------------------------------------------------
// Kernel C2: scatter gts[b, lsa] = labels[b, idx]; pos/neg ratio per image
// ---------------------------------------------------------------------------
__global__ void gts_kernel(const int* __restrict__ labels,
                           const int* __restrict__ lsa,
                           const int* __restrict__ idxArr,
                           int* __restrict__ gts,
                           float* __restrict__ redarr)
{
    __shared__ int cnt[256];
    int b = blockIdx.x, tid = threadIdx.x;
    for (int p = tid; p < P_N; p += 256) gts[b * P_N + p] = 0;
    __syncthreads();
    if (tid < G_N)
        gts[b * P_N + lsa[b * G_N + tid]] = labels[b * G_N + idxArr[b * G_N + tid]];
    __syncthreads();
    int c = 0;
    for (int p = tid; p < P_N; p += 256) c += (gts[b * P_N + p] != 0);
    cnt[tid] = c;
    __syncthreads();
    for (int s = 128; s > 0; s >>= 1) {
        if (tid < s) cnt[tid] += cnt[tid + s];
        __syncthreads();
    }
    if (tid == 0) {
        int pos = cnt[0];
        redarr[b] = ((float)pos / (float)(P_N - pos)) * 1.25f;   // red * BG_SCALE
    }
}

// ---------------------------------------------------------------------------
// Kernel D: focal loss elementwise over [B,P,C], per-block partial sums
// ---------------------------------------------------------------------------
__global__ void focal_kernel(const float* __restrict__ pc,
                             const int* __restrict__ gts,
                             const float* __restrict__ redarr,
                             float* __restrict__ partials)
{
    __shared__ float sm[256];
    const long long total  = (long long)B_N * P_N * C_N;
    const long long stride = (long long)gridDim.x * blockDim.x;
    float acc = 0.f;
    for (long long e = (long long)blockIdx.x * blockDim.x + threadIdx.x;
         e < total; e += stride) {
        float x = pc[e];
        __builtin_prefetch(pc + e + stride, 0, 0);   // global_prefetch_b8
        int c        = (int)(e & (C_N - 1));
        long long bp = e >> 9;                        // C_N == 512
        int g = gts[bp];
        float t    = (c == g) ? 1.f : 0.f;
        float ce   = fmaxf(x, 0.f) - x * t + log1pf(expf(-fabsf(x)));
        float prob = 1.f / (1.f + expf(-x));
        float pt   = prob * t + (1.f - prob) * (1.f - t);
        float at   = 0.75f * t + 0.25f * (1.f - t);
        float om   = 1.f - pt;
        float fl   = at * ce * om * om;
        if (g == 0) fl *= redarr[(int)(bp / P_N)];
        acc += fl;
    }
    sm[threadIdx.x] = acc;
    __syncthreads();
    for (int s = 128; s > 0; s >>= 1) {
        if (threadIdx.x < s) sm[threadIdx.x] += sm[threadIdx.x + s];
        __syncthreads();
    }
    if (threadIdx.x == 0) partials[blockIdx.x] = sm[0];
}

// ---------------------------------------------------------------------------
// Kernel E: deterministic final reduction of 4096 partials via
// V_WMMA_F32_16X16X4_F32 (C += A(16x4) * ones(4x16)); one wave32.
// A-matrix VGPR layout: lanes 0-15 hold M=lane, K0/K1 in the 2 VGPRs;
// lanes 16-31 hold M=lane-16, K2/K3.
// ---------------------------------------------------------------------------
__global__ void reduce_wmma_kernel(const float* __restrict__ partials,
                                   float* __restrict__ out)
{
    __shared__ float sm[32];
    int lane = threadIdx.x;
    int Mr = lane & 15;
    int Kb = (lane >> 4) * 2;
    v8f acc = {0.f, 0.f, 0.f, 0.f, 0.f, 0.f, 0.f, 0.f};
    v2f ones; ones.x = 1.f; ones.y = 1.f;
    for (int t = 0; t < NBLK / 64; ++t) {
        v2f a;
        a.x = partials[t * 64 + Kb * 16 + Mr];
        a.y = partials[t * 64 + (Kb + 1) * 16 + Mr];
        acc = __builtin_amdgcn_wmma_f32_16x16x4_f32(
            /*neg_a=*/false, a, /*neg_b=*/false, ones,
            /*c_mod=*/(short)0, acc, /*reuse_a=*/false, /*reuse_b=*/false);
    }
    // column 0 of C: lane 0 holds M=0..7 partial rows, lane 16 holds M=8..15
    float s = acc[0] + acc[1] + acc[2] + acc[3] + acc[4] + acc[5] + acc[6] + acc[7];
    sm[lane] = s;
    __syncthreads();
    if (lane == 0) out[1] = (sm[0] + sm[16]) / (float)B_N;
}

// ---------------------------------------------------------------------------
extern "C" void kernel_launch(void* const* d_in, const int* in_sizes, int n_in,
                              void* d_out, int out_size, void* d_ws, size_t ws_size,
                              hipStream_t stream)
{
    const float* pred_boxes   = (const float*)d_in[0];   // [B,P,4] f32
    const float* pred_classes = (const float*)d_in[1];   // [B,P,C] f32
    const float* boxes        = (const float*)d_in[2];   // [B,G,4] f32
    const int*   labels       = (const int*)d_in[3];     // [B,G]  i32
    float* out = (float*)d_out;   // [0]=box_loss, [1]=class_loss, [2..]=pm[B,G,4]

    char* ws = (char*)d_ws;
    size_t off = 0;
    double* cost = (double*)(ws + off); off += (size_t)B_N * G_N * P_N * sizeof(double);
    int* lsa     = (int*)(ws + off);    off += (size_t)B_N * G_N * sizeof(int);
    int* idxA    = (int*)(ws + off);    off += (size_t)B_N * G_N * sizeof(int);
    int* gts     = (int*)(ws + off);    off += (size_t)B_N * P_N * sizeof(int);
    float* redarr = (float*)(ws + off); off += 256;                     // B floats, padded
    float* partials = (float*)(ws + off); off += (size_t)NBLK * sizeof(float);

    const int costTotal = B_N * G_N * P_N;
    cost_kernel<<<(costTotal + 255) / 256, 256, 0, stream>>>(pred_boxes, boxes, cost);
    hungarian_kernel<<<B_N, 256, 0, stream>>>(cost, lsa, idxA);
    boxloss_pm_kernel<<<B_N, G_N, 0, stream>>>(pred_boxes, boxes, lsa, idxA, out);
    gts_kernel<<<B_N, 256, 0, stream>>>(labels, lsa, idxA, gts, redarr);
    focal_kernel<<<NBLK, 256, 0, stream>>>(pred_classes, gts, redarr, partials);
    reduce_wmma_kernel<<<1, 32, 0, stream>>>(partials, out);
}